// PointPillarWhat2keep_54185307406439
// MI455X (gfx1250) — compile-verified
//
#include <hip/hip_runtime.h>

typedef __attribute__((ext_vector_type(2))) float v2f;
typedef __attribute__((ext_vector_type(8))) float v8f;

#define WMMA_F32(a, b, c) \
  __builtin_amdgcn_wmma_f32_16x16x4_f32(false, (a), false, (b), (short)0, (c), false, false)

static constexpr int kAgent = 6, kWin = 7, kDim = 256, kHeads = 8, kDh = 32;
static constexpr int kNTok = kAgent * kWin * kWin;   // 294
static constexpr int kNT   = 19;                     // ceil(294/16)
static constexpr int kNPad = kNT * 16;               // 304
static constexpr int kX = 16, kY = 16;
static constexpr int kNWin = kX * kY;                // 256 windows
static constexpr int kQKVCols = 3 * kDim;            // 768
static constexpr float kScale = 0.17677669529663687f; // 32^-0.5
static constexpr size_t kHeadElems = (size_t)kNWin * kHeads * kNTok * kDh; // q/k/v each

// -------------------------------------------------------------------------
// Kernel 1: QKV projection. One wave computes a 16x64 slab (1 row-tile x 4
// col-tiles) of (294 x 768) per window: A fragment reused 4x per k-step,
// 4 x V_WMMA_F32_16X16X4_F32 per step. A rows gathered through the
// (b,l,x,y,w1,w2,d) -> (bxy, lw1w2, d) permutation; splits q/k/v per head.
// -------------------------------------------------------------------------
__global__ __launch_bounds__(32) void qkv_kernel(const float* __restrict__ x,
                                                 const float* __restrict__ w_qkv,
                                                 float* __restrict__ q,
                                                 float* __restrict__ k,
                                                 float* __restrict__ v) {
  int bid = blockIdx.x;
  int ct  = bid % 12;                 // 12 slabs of 64 cols
  int rt  = (bid / 12) % kNT;
  int win = bid / (12 * kNT);
  int wx = win / kY, wy = win % kY;
  int lane = threadIdx.x;
  int lm = lane & 15, hl = lane >> 4;

  int n  = rt * 16 + lm;
  int nc = (n < kNTok) ? n : (kNTok - 1);
  int l = nc / 49, rem = nc % 49, w1 = rem / 7, w2 = rem % 7;
  const float* arow =
      x + ((size_t)((((l * kX + wx) * kY + wy) * kWin + w1) * kWin + w2)) * kDim;
  int cbase = ct * 64 + lm;

  // Prefetch parts of the gathered 1KB A rows early (global_prefetch_b8):
  // lane l hits its own row lm at byte 32*l, so each row gets only 2 spots.
  __builtin_prefetch(arow + lane * 8, 0, 1);

  v8f acc[4] = {};
  for (int kb = 0; kb < kDim; kb += 4) {
    int ka = kb + hl * 2;
    v2f a;
    a.x = arow[ka];
    a.y = arow[ka + 1];
    const float* brow0 = w_qkv + (size_t)ka * kQKVCols + cbase;
    const float* brow1 = brow0 + kQKVCols;
#pragma unroll
    for (int t = 0; t < 4; ++t) {
      v2f b;
      b.x = brow0[16 * t];
      b.y = brow1[16 * t];
      acc[t] = WMMA_F32(a, b, acc[t]);
    }
  }

#pragma unroll
  for (int t = 0; t < 4; ++t) {
    int c = cbase + 16 * t;
    int which = c >> 8;               // 0=q, 1=k, 2=v
    int cc = c & 255;
    int h = cc >> 5, d = cc & 31;
#pragma unroll
    for (int r = 0; r < 8; ++r) {
      int row = rt * 16 + r + hl * 8;
      if (row >= kNTok) continue;
      float val = acc[t][r];
      size_t off = (((size_t)win * kHeads + h) * kNTok + row) * kDh + d;
      if (which == 0)      q[off] = val * kScale;
      else if (which == 1) k[off] = val;
      else                 v[off] = val;
    }
  }
}

// -------------------------------------------------------------------------
// Kernel 2: fused attention per (window, head). K,V staged in LDS with b128
// loads (zero pad to 304 rows). Each wave handles 16-row blocks: QK^T via
// WMMA, +bias gather, pad-mask, softmax (shfl_xor cross-half reduce), then
// P@V via WMMA, scaled by 1/rowsum at store.
// -------------------------------------------------------------------------
__global__ __launch_bounds__(128) void attn_kernel(const float* __restrict__ q,
                                                   const float* __restrict__ kbuf,
                                                   const float* __restrict__ vbuf,
                                                   const float* __restrict__ bias_table,
                                                   const int* __restrict__ rel_index,
                                                   float* __restrict__ o) {
  __shared__ float4 kls4[kNPad * kDh / 4];   // 38 912 B
  __shared__ float4 vls4[kNPad * kDh / 4];   // 38 912 B
  __shared__ float  pls[4][16 * kNPad];      // 77 824 B (per-wave score rows)
  __shared__ float  rsum[4][16];             //    256 B
  float* kls = (float*)kls4;
  float* vls = (float*)vls4;

  int bh = blockIdx.x;
  int win = bh / kHeads, h = bh % kHeads;
  const float*  qg  = q + (size_t)bh * kNTok * kDh;
  const float4* kg4 = (const float4*)(kbuf + (size_t)bh * kNTok * kDh);
  const float4* vg4 = (const float4*)(vbuf + (size_t)bh * kNTok * kDh);

  for (int idx = threadIdx.x; idx < kNPad * kDh / 4; idx += blockDim.x) {
    bool inb = idx < kNTok * kDh / 4;
    float4 zero = make_float4(0.f, 0.f, 0.f, 0.f);
    kls4[idx] = inb ? kg4[idx] : zero;
    vls4[idx] = inb ? vg4[idx] : zero;
  }
  __syncthreads();

  int wid = threadIdx.x >> 5;
  int lane = threadIdx.x & 31;
  int lm = lane & 15, hl = lane >> 4;
  float* P = pls[wid];

  for (int it = 0; it < 5; ++it) {     // fixed trip count: barriers stay uniform
    int rb = it * 4 + wid;
    bool active = rb < kNT;

    if (active) {
      // Q fragments for this 16-row block (A-matrix layout, kept in VGPRs)
      int i0 = rb * 16 + lm;
      int iq = (i0 < kNTok) ? i0 : (kNTok - 1);
      const float* qrow = qg + (size_t)iq * kDh;
      v2f qf[8];
#pragma unroll
      for (int ks = 0; ks < 8; ++ks) {
        int c = ks * 4 + hl * 2;
        qf[ks].x = qrow[c];
        qf[ks].y = qrow[c + 1];
      }
      // scores: 19 j-tiles, K=32 (8 WMMA each); bias-add + mask at LDS store
      for (int jt = 0; jt < kNT; ++jt) {
        v8f c = {};
        int j = jt * 16 + lm;
#pragma unroll
        for (int ks = 0; ks < 8; ++ks) {
          int kc = ks * 4 + hl * 2;
          v2f b;
          b.x = kls[j * kDh + kc];
          b.y = kls[j * kDh + kc + 1];
          c = WMMA_F32(qf[ks], b, c);
        }
#pragma unroll
        for (int r = 0; r < 8; ++r) {
          int rl = r + hl * 8;
          int i = rb * 16 + rl;
          float sv = -1e30f;
          if (i < kNTok && j < kNTok)
            sv = c[r] + bias_table[(size_t)rel_index[i * kNTok + j] * kHeads + h];
          P[rl * kNPad + j] = sv;
        }
      }
    }
    __syncthreads();

    if (active) {
      // softmax: lane owns row lm, half hl owns 152 of the 304 columns
      const int cols = kNPad / 2;
      float* prw = P + lm * kNPad + hl * cols;
      float m = -1e30f;
      for (int c = 0; c < cols; ++c) m = fmaxf(m, prw[c]);
      m = fmaxf(m, __shfl_xor(m, 16, 32));
      float s = 0.f;
      for (int c = 0; c < cols; ++c) {
        float e = expf(prw[c] - m);
        prw[c] = e;
        s += e;
      }
      s += __shfl_xor(s, 16, 32);
      if (hl == 0) rsum[wid][lm] = 1.0f / s;
    }
    __syncthreads();

    if (active) {
      // out_i = P (16x304) @ V (304x32); two 16-col blocks, 76 WMMA each
#pragma unroll
      for (int cb = 0; cb < 2; ++cb) {
        v8f oc = {};
        int d = cb * 16 + lm;
        for (int kt = 0; kt < kNPad; kt += 4) {
          int kc = kt + hl * 2;
          v2f a, b;
          a.x = P[lm * kNPad + kc];
          a.y = P[lm * kNPad + kc + 1];
          b.x = vls[kc * kDh + d];
          b.y = vls[(kc + 1) * kDh + d];
          oc = WMMA_F32(a, b, oc);
        }
#pragma unroll
        for (int r = 0; r < 8; ++r) {
          int rl = r + hl * 8;
          int row = rb * 16 + rl;
          if (row >= kNTok) continue;
          o[((size_t)win * kNTok + row) * kDim + h * kDh + d] = oc[r] * rsum[wid][rl];
        }
      }
    }
    __syncthreads();
  }
}

// -------------------------------------------------------------------------
// Kernel 3: output projection (294x256)@(256x256) per window, 16x64 slab per
// wave, inverse permutation back to (b,l,x,y,w1,w2,d) on store.
// -------------------------------------------------------------------------
__global__ __launch_bounds__(32) void proj_kernel(const float* __restrict__ o,
                                                  const float* __restrict__ w_out,
                                                  float* __restrict__ out) {
  int bid = blockIdx.x;
  int ct  = bid % 4;                  // 4 slabs of 64 cols
  int rt  = (bid / 4) % kNT;
  int win = bid / (4 * kNT);
  int wx = win / kY, wy = win % kY;
  int lane = threadIdx.x;
  int lm = lane & 15, hl = lane >> 4;

  int n  = rt * 16 + lm;
  int nc = (n < kNTok) ? n : (kNTok - 1);
  const float* arow = o + ((size_t)win * kNTok + nc) * kDim;
  int cbase = ct * 64 + lm;

  __builtin_prefetch(arow + lane * 8, 0, 1);

  v8f acc[4] = {};
  for (int kb = 0; kb < kDim; kb += 4) {
    int ka = kb + hl * 2;
    v2f a;
    a.x = arow[ka];
    a.y = arow[ka + 1];
    const float* brow0 = w_out + (size_t)ka * kDim + cbase;
    const float* brow1 = brow0 + kDim;
#pragma unroll
    for (int t = 0; t < 4; ++t) {
      v2f b;
      b.x = brow0[16 * t];
      b.y = brow1[16 * t];
      acc[t] = WMMA_F32(a, b, acc[t]);
    }
  }

#pragma unroll
  for (int r = 0; r < 8; ++r) {
    int row = rt * 16 + r + hl * 8;
    if (row >= kNTok) continue;
    int l = row / 49, rem = row % 49, w1 = rem / 7, w2 = rem % 7;
    size_t base =
        ((size_t)((((l * kX + wx) * kY + wy) * kWin + w1) * kWin + w2)) * kDim;
#pragma unroll
    for (int t = 0; t < 4; ++t) {
      out[base + cbase + 16 * t] = acc[t][r];
    }
  }
}

extern "C" void kernel_launch(void* const* d_in, const int* in_sizes, int n_in,
                              void* d_out, int out_size, void* d_ws, size_t ws_size,
                              hipStream_t stream) {
  const float* x          = (const float*)d_in[0];
  const float* w_qkv      = (const float*)d_in[1];
  const float* w_out      = (const float*)d_in[2];
  const float* bias_table = (const float*)d_in[3];
  const int*   rel_index  = (const int*)d_in[4];
  float* out = (float*)d_out;

  float* q = (float*)d_ws;          // each q/k/v: 256*8*294*32 floats
  float* k = q + kHeadElems;
  float* v = k + kHeadElems;
  float* o = v + kHeadElems;        // 256*294*256 floats

  qkv_kernel<<<kNWin * kNT * 12, 32, 0, stream>>>(x, w_qkv, q, k, v);
  attn_kernel<<<kNWin * kHeads, 128, 0, stream>>>(q, k, v, bias_table, rel_index, o);
  proj_kernel<<<kNWin * kNT * 4, 32, 0, stream>>>(o, w_out, out);
}